// PROFIT_16776142258476
// MI455X (gfx1250) — compile-verified
//
#include <hip/hip_runtime.h>

// GCN forward for MI455X (gfx1250), wave32, WMMA f32 16x16x4 matrix path.
//
// d_in: x[100000*128] f32, edge_index[2*1600000] i32, q_emb[50000*128] f32,
//       W1[128*128], b1[128], W2[128*128], b2[128], Wm[128*128], bm[128]
// d_out: ques[50000*128] ++ h2[100000*128]  (f32)
// d_ws : dinv[N] + H[N*128]  (~52 MB)

typedef __attribute__((ext_vector_type(2))) float v2f;
typedef __attribute__((ext_vector_type(8))) float v8f;

#define DDIM 128
#define LDSP 132   // padded LDS row stride (floats): avoids bank conflicts

// ---- degree / norm kernels -------------------------------------------------

__global__ void gcn_init_deg(float* __restrict__ deg, int n) {
  int i = blockIdx.x * blockDim.x + threadIdx.x;
  if (i < n) deg[i] = 1.0f;                  // self-loop contributes 1
}

__global__ void gcn_count_deg(const int* __restrict__ col, float* __restrict__ deg, int E) {
  int e = blockIdx.x * blockDim.x + threadIdx.x;
  if (e < E) unsafeAtomicAdd(&deg[col[e]], 1.0f);
}

__global__ void gcn_rsqrt_deg(float* __restrict__ deg, int n) {
  int i = blockIdx.x * blockDim.x + threadIdx.x;
  if (i < n) deg[i] = rsqrtf(deg[i]);        // deg >= 1 always (self-loop)
}

// ---- WMMA fp32 GEMM: Out[M,128] = op(A)[M,128] @ W[128,128] (+bias, relu) --
// relu_in != 0  : apply ReLU to A elements while staging to LDS
// bias != null  : epilogue Out = relu(acc + bias[n])

__global__ void __launch_bounds__(256)
gcn_gemm_wmma_f32(const float* __restrict__ A, const float* __restrict__ W,
                  float* __restrict__ Out, int M, int relu_in,
                  const float* __restrict__ bias) {
  __shared__ float Alds[16 * LDSP];
  __shared__ float Blds[DDIM * LDSP];

  const int tid = threadIdx.x;
  const int m0  = blockIdx.x * 16;

  // Stage W (128x128) -> LDS, coalesced, 64 floats/thread
  for (int i = 0; i < (DDIM * DDIM) / 256; ++i) {
    int flat = i * 256 + tid;
    int r = flat >> 7, c = flat & 127;
    Blds[r * LDSP + c] = W[flat];
  }
  // Stage A tile (16x128) -> LDS, 8 floats/thread, optional ReLU on load
  for (int i = 0; i < (16 * DDIM) / 256; ++i) {
    int flat = i * 256 + tid;
    int r = flat >> 7, c = flat & 127;
    int m = m0 + r;
    float v = (m < M) ? A[(long)m * DDIM + c] : 0.0f;
    if (relu_in) v = v > 0.0f ? v : 0.0f;
    Alds[r * LDSP + c] = v;
  }
  __syncthreads();

  const int wave = tid >> 5;        // 8 waves -> 8 x 16-column slices
  const int lane = tid & 31;
  const int half = lane >> 4;       // K/M sub-tile selector per ISA layout
  const int lm   = lane & 15;
  const int n0   = wave * 16;

  v8f acc = {};                     // 16x16 f32 C/D tile (8 VGPRs)
  #pragma unroll
  for (int k0 = 0; k0 < DDIM; k0 += 4) {
    const int k = k0 + 2 * half;
    v2f af, bf;
    // A 16x4 fragment: lane -> M = lm, VGPR v -> K = k + v
    af.x = Alds[lm * LDSP + k];
    af.y = Alds[lm * LDSP + k + 1];
    // B 4x16 fragment: lane -> N = lm, VGPR v -> K = k + v
    bf.x = Blds[k * LDSP + n0 + lm];
    bf.y = Blds[(k + 1) * LDSP + n0 + lm];
    acc = __builtin_amdgcn_wmma_f32_16x16x4_f32(
        /*neg_a=*/false, af, /*neg_b=*/false, bf,
        /*c_mod=*/(short)0, acc, /*reuse_a=*/false, /*reuse_b=*/false);
  }

  const int n = n0 + lm;
  if (bias) {
    const float bv = bias[n];
    #pragma unroll
    for (int r = 0; r < 8; ++r) {
      int m = m0 + r + 8 * half;    // C/D layout: VGPR r -> M = r + 8*half
      if (m < M) {
        float v = acc[r] + bv;
        Out[(long)m * DDIM + n] = v > 0.0f ? v : 0.0f;
      }
    }
  } else {
    #pragma unroll
    for (int r = 0; r < 8; ++r) {
      int m = m0 + r + 8 * half;
      if (m < M) Out[(long)m * DDIM + n] = acc[r];
    }
  }
}

// ---- aggregation -----------------------------------------------------------
// agg[n,:] = H[n,:] * dinv[n]^2 + b[:]   (self-loop message + bias, one shot)
__global__ void gcn_agg_init(const float* __restrict__ H, const float* __restrict__ dinv,
                             const float* __restrict__ b, float* __restrict__ agg,
                             int total) {
  int i = blockIdx.x * blockDim.x + threadIdx.x;
  if (i < total) {
    int node = i >> 7, d = i & 127;
    float s = dinv[node];
    agg[i] = H[i] * s * s + b[d];
  }
}

// agg[col,:] += H[row,:] * dinv[row]*dinv[col]   (2 edges per 256-thread block)
__global__ void gcn_scatter(const float* __restrict__ H, const float* __restrict__ dinv,
                            const int* __restrict__ row, const int* __restrict__ col,
                            float* __restrict__ agg, int E) {
  int e = blockIdx.x * 2 + (threadIdx.x >> 7);
  if (e >= E) return;
  int d = threadIdx.x & 127;
  int r = row[e], c = col[e];
  float norm = dinv[r] * dinv[c];
  unsafeAtomicAdd(&agg[(long)c * DDIM + d], H[(long)r * DDIM + d] * norm);
}

// ---- launch ----------------------------------------------------------------

extern "C" void kernel_launch(void* const* d_in, const int* in_sizes, int n_in,
                              void* d_out, int out_size, void* d_ws, size_t ws_size,
                              hipStream_t stream) {
  const float* x    = (const float*)d_in[0];
  const int*   edge = (const int*)  d_in[1];
  const float* q    = (const float*)d_in[2];
  const float* W1   = (const float*)d_in[3];
  const float* b1   = (const float*)d_in[4];
  const float* W2   = (const float*)d_in[5];
  const float* b2   = (const float*)d_in[6];
  const float* Wm   = (const float*)d_in[7];
  const float* bm   = (const float*)d_in[8];

  const int N  = in_sizes[0] / DDIM;   // 100000
  const int E  = in_sizes[1] / 2;      // 1600000
  const int NQ = in_sizes[2] / DDIM;   // 50000

  float* outQ  = (float*)d_out;                    // [NQ,128]
  float* outH2 = outQ + (long)NQ * DDIM;           // [N,128] final h2 + agg scratch

  float* dinv = (float*)d_ws;                      // [N]
  float* H    = dinv + ((N + 1023) / 1024) * 1024; // [N,128] GEMM temp

  const int* row = edge;        // messages flow row -> col
  const int* col = edge + E;

  const int T = 256;
  const int nb_nodes = (N + T - 1) / T;
  const int nb_edges = (E + T - 1) / T;
  const int nb_feat  = (N * DDIM + T - 1) / T;
  const int nb_scat  = (E + 1) / 2;

  // D^{-1/2}
  gcn_init_deg <<<nb_nodes, T, 0, stream>>>(dinv, N);
  gcn_count_deg<<<nb_edges, T, 0, stream>>>(col, dinv, E);
  gcn_rsqrt_deg<<<nb_nodes, T, 0, stream>>>(dinv, N);

  // conv1: H = x@W1 ; agg(in outH2) = H*dinv^2 + b1 ; scatter edges
  gcn_gemm_wmma_f32<<<(N + 15) / 16, T, 0, stream>>>(x, W1, H, N, 0, nullptr);
  gcn_agg_init<<<nb_feat, T, 0, stream>>>(H, dinv, b1, outH2, N * DDIM);
  gcn_scatter <<<nb_scat, T, 0, stream>>>(H, dinv, row, col, outH2, E);

  // conv2: H = relu(agg)@W2 (ReLU fused on A-load) ; agg = H*dinv^2 + b2 ; scatter
  gcn_gemm_wmma_f32<<<(N + 15) / 16, T, 0, stream>>>(outH2, W2, H, N, 1, nullptr);
  gcn_agg_init<<<nb_feat, T, 0, stream>>>(H, dinv, b2, outH2, N * DDIM);
  gcn_scatter <<<nb_scat, T, 0, stream>>>(H, dinv, row, col, outH2, E);

  // ques = relu(q@Wm + bm), fused epilogue, straight into d_out
  gcn_gemm_wmma_f32<<<(NQ + 15) / 16, T, 0, stream>>>(q, Wm, outQ, NQ, 0, bm);
}